// CUDAGaussianModel_35940286332983
// MI455X (gfx1250) — compile-verified
//
#include <hip/hip_runtime.h>

typedef float v2f __attribute__((ext_vector_type(2)));
typedef float v8f __attribute__((ext_vector_type(8)));

#define N_GAUSS   4096
#define M_POINTS  65536
#define GSTRIDE   16          // floats per gaussian coefficient row (12 used, padded)
#define NTILES    (N_GAUSS / 16)
#define WAVES_PER_BLOCK 8     // 256 threads, wave32
#define PTS_PER_BLOCK   (WAVES_PER_BLOCK * 16)

// -0.5 * log2(e): fold exp(-0.5 q) -> exp2(kS * q)
#define KSCALE (-0.72134752044448169f)

// ---------------------------------------------------------------------------
// Prep: per-gaussian coefficient vector g[n][0..11], pre-scaled by -0.5*log2(e)
// and with log2(intensity) folded into the constant slot.
// q(x) = x^T A x - 2 x.b + c ;  t = kS*q + log2(i) ;  contribution = exp2(t)
// ---------------------------------------------------------------------------
__global__ __launch_bounds__(256) void gauss_prep(
    const float* __restrict__ positions,   // [N,3]
    const float* __restrict__ scales,      // [N,3]
    const float* __restrict__ rotations,   // [N,4]
    const float* __restrict__ intensities, // [N]
    float* __restrict__ g)                 // [N,GSTRIDE]
{
    int n = blockIdx.x * blockDim.x + threadIdx.x;
    if (n >= N_GAUSS) return;

    float qw = rotations[n*4+0], qx = rotations[n*4+1];
    float qy = rotations[n*4+2], qz = rotations[n*4+3];
    float nrm = sqrtf(qw*qw + qx*qx + qy*qy + qz*qz) + 1e-8f;
    qw /= nrm; qx /= nrm; qy /= nrm; qz /= nrm;

    float R00 = 1.f - 2.f*(qy*qy + qz*qz), R01 = 2.f*(qx*qy - qz*qw), R02 = 2.f*(qx*qz + qy*qw);
    float R10 = 2.f*(qx*qy + qz*qw), R11 = 1.f - 2.f*(qx*qx + qz*qz), R12 = 2.f*(qy*qz - qx*qw);
    float R20 = 2.f*(qx*qz - qy*qw), R21 = 2.f*(qy*qz + qx*qw), R22 = 1.f - 2.f*(qx*qx + qy*qy);

    float sx = fabsf(scales[n*3+0]) + 1e-6f;
    float sy = fabsf(scales[n*3+1]) + 1e-6f;
    float sz = fabsf(scales[n*3+2]) + 1e-6f;
    float i0 = 1.f/(sx*sx), i1 = 1.f/(sy*sy), i2 = 1.f/(sz*sz);

    // A = R diag(i) R^T (symmetric)
    float A00 = R00*i0*R00 + R01*i1*R01 + R02*i2*R02;
    float A01 = R00*i0*R10 + R01*i1*R11 + R02*i2*R12;
    float A02 = R00*i0*R20 + R01*i1*R21 + R02*i2*R22;
    float A11 = R10*i0*R10 + R11*i1*R11 + R12*i2*R12;
    float A12 = R10*i0*R20 + R11*i1*R21 + R12*i2*R22;
    float A22 = R20*i0*R20 + R21*i1*R21 + R22*i2*R22;

    float px = positions[n*3+0], py = positions[n*3+1], pz = positions[n*3+2];
    float b0 = A00*px + A01*py + A02*pz;
    float b1 = A01*px + A11*py + A12*pz;
    float b2 = A02*px + A12*py + A22*pz;
    float c  = b0*px + b1*py + b2*pz;

    float* o = g + n * GSTRIDE;
    o[0] = KSCALE * A00;
    o[1] = KSCALE * A11;
    o[2] = KSCALE * A22;
    o[3] = 2.f * KSCALE * A01;
    o[4] = 2.f * KSCALE * A02;
    o[5] = 2.f * KSCALE * A12;
    o[6] = -2.f * KSCALE * b0;
    o[7] = -2.f * KSCALE * b1;
    o[8] = -2.f * KSCALE * b2;
    o[9] = KSCALE * c + log2f(fmaxf(intensities[n], 1e-38f));
    o[10] = 0.f; o[11] = 0.f; o[12] = 0.f; o[13] = 0.f; o[14] = 0.f; o[15] = 0.f;
}

// ---------------------------------------------------------------------------
// Main: each wave owns 16 points (A tile, loop-invariant fragments), iterates
// over 256 gaussian tiles of 16. Per tile: 3x v_wmma_f32_16x16x4_f32 (K=12),
// then 8x v_exp_f32 + add per lane. Column reduction via 16-lane shfl_xor.
// ---------------------------------------------------------------------------
__global__ __launch_bounds__(256) void gauss_eval(
    const float* __restrict__ pts,  // [M,3]
    const float* __restrict__ g,    // [N,GSTRIDE]
    float* __restrict__ out)        // [M]
{
    __shared__ float fsh[WAVES_PER_BLOCK * 16 * GSTRIDE];

    int lane  = threadIdx.x & 31;
    int wid   = threadIdx.x >> 5;
    int mbase = (blockIdx.x * WAVES_PER_BLOCK + wid) * 16;

    float* fw = &fsh[wid * 16 * GSTRIDE];

    // lanes 0..15 build the 16 point-feature rows into LDS (same-wave, in-order)
    if (lane < 16) {
        const float* p = pts + (size_t)(mbase + lane) * 3;
        float x = p[0], y = p[1], z = p[2];
        float* fr = fw + lane * GSTRIDE;
        fr[0] = x*x; fr[1] = y*y; fr[2] = z*z;
        fr[3] = x*y; fr[4] = x*z; fr[5] = y*z;
        fr[6] = x;   fr[7] = y;   fr[8] = z;
        fr[9] = 1.f; fr[10] = 0.f; fr[11] = 0.f;
    }

    // ISA 32-bit A 16x4 layout: lanes 0-15 hold M=0..15 K={4k+0,4k+1},
    // lanes 16-31 hold M=0..15 K={4k+2,4k+3}. B (4x16) mirrors with N across lanes.
    int row   = lane & 15;
    int khalf = (lane >> 4) << 1;  // 0 or 2

    v2f a0 = *(const v2f*)(fw + row * GSTRIDE + 0 + khalf);
    v2f a1 = *(const v2f*)(fw + row * GSTRIDE + 4 + khalf);
    v2f a2 = *(const v2f*)(fw + row * GSTRIDE + 8 + khalf);

    float acc[8] = {0.f, 0.f, 0.f, 0.f, 0.f, 0.f, 0.f, 0.f};

    for (int j = 0; j < NTILES; ++j) {
        const float* grow = g + (size_t)(j * 16 + row) * GSTRIDE;
        v2f b0 = *(const v2f*)(grow + 0 + khalf);
        v2f b1 = *(const v2f*)(grow + 4 + khalf);
        v2f b2 = *(const v2f*)(grow + 8 + khalf);

        v8f c = {};
        c = __builtin_amdgcn_wmma_f32_16x16x4_f32(false, a0, false, b0, (short)0, c, false, false);
        c = __builtin_amdgcn_wmma_f32_16x16x4_f32(false, a1, false, b1, (short)0, c, false, false);
        c = __builtin_amdgcn_wmma_f32_16x16x4_f32(false, a2, false, b2, (short)0, c, false, false);

#pragma unroll
        for (int i = 0; i < 8; ++i)
            acc[i] += __builtin_amdgcn_exp2f(c[i]);
    }

    // Reduce the 16 column-partials within each half-wave (butterfly keeps
    // lanes inside their 16-lane group for masks 1,2,4,8).
#pragma unroll
    for (int i = 0; i < 8; ++i) {
        float v = acc[i];
        v += __shfl_xor(v, 1);
        v += __shfl_xor(v, 2);
        v += __shfl_xor(v, 4);
        v += __shfl_xor(v, 8);
        if (row == 0)  // lane 0 -> rows 0..7, lane 16 -> rows 8..15
            out[mbase + i + ((lane >> 4) << 3)] = v;
    }
}

extern "C" void kernel_launch(void* const* d_in, const int* in_sizes, int n_in,
                              void* d_out, int out_size, void* d_ws, size_t ws_size,
                              hipStream_t stream) {
    const float* pts   = (const float*)d_in[0];
    const float* pos   = (const float*)d_in[1];
    const float* scl   = (const float*)d_in[2];
    const float* rot   = (const float*)d_in[3];
    const float* inten = (const float*)d_in[4];

    float* g   = (float*)d_ws;   // N_GAUSS * GSTRIDE * 4 = 256 KB
    float* out = (float*)d_out;

    gauss_prep<<<N_GAUSS / 256, 256, 0, stream>>>(pos, scl, rot, inten, g);
    gauss_eval<<<M_POINTS / PTS_PER_BLOCK, 256, 0, stream>>>(pts, g, out);
}